// DeepEquilibriumLayer_43336220017043
// MI455X (gfx1250) — compile-verified
//
#include <hip/hip_runtime.h>
#include <math.h>

typedef float v2f __attribute__((ext_vector_type(2)));
typedef float v8f __attribute__((ext_vector_type(8)));

#define HIDDEN        128
#define ROWS_PER_WAVE 16
#define WAVES_PER_BLK 8
#define THREADS       (WAVES_PER_BLK * 32)
#define LDS_STRIDE    132                         // padded floats per row (132 % 64 == 4)
#define W_LDS_FLOATS  (HIDDEN * LDS_STRIDE)       // 16896 floats = 67584 B
#define Z_LDS_FLOATS  (ROWS_PER_WAVE * LDS_STRIDE)// 2112 floats  =  8448 B
#define MAX_ITER      50
#define CONV_EPS      1e-7f

__global__ __launch_bounds__(THREADS, 4)
void deq_f32_wmma_kernel(const float* __restrict__ x, const float* __restrict__ W,
                         const float* __restrict__ b, float* __restrict__ out)
{
    extern __shared__ float lds[];
    float* wsh = lds;                                           // [128][132] padded W
    const int tid   = threadIdx.x;
    const int lane  = tid & 31;
    const int wave  = tid >> 5;
    const int lmod  = lane & 15;
    const int lhalf = lane >> 4;
    float* zsh = lds + W_LDS_FLOATS + wave * Z_LDS_FLOATS;      // per-wave z staging

    // ---- cooperative load of W into padded LDS (once) ----
    {
        const float4* Wv = (const float4*)W;                    // [128][32] float4
        const int c4 = tid & 31;
        const int r0 = tid >> 5;
        #pragma unroll
        for (int i = 0; i < 16; ++i) {
            const int row = r0 + i * 8;
            float4 v = Wv[row * 32 + c4];
            *(float4*)(wsh + row * LDS_STRIDE + c4 * 4) = v;
        }
    }
    __syncthreads();   // only barrier; everything after is wave-private

    const long row0 = (long)blockIdx.x * (ROWS_PER_WAVE * WAVES_PER_BLK)
                    + (long)wave * ROWS_PER_WAVE;

    // ---- xb = x + b in C/D layout registers; z1 = tanh(xb) stored to LDS ----
    v8f xb[8];
    #pragma unroll
    for (int n = 0; n < 8; ++n) {
        const int col  = n * 16 + lmod;
        const float bv = b[col];
        v8f v;
        #pragma unroll
        for (int m = 0; m < 8; ++m) {
            const int row = m + 8 * lhalf;                      // D-layout row
            v[m] = x[(row0 + row) * HIDDEN + col] + bv;
        }
        xb[n] = v;
        #pragma unroll
        for (int m = 0; m < 8; ++m)
            zsh[(m + 8 * lhalf) * LDS_STRIDE + col] = tanhf(v[m]);
    }
    asm volatile("s_wait_dscnt 0x0" ::: "memory");

    // ---- fixed-point iterations 2..50 with per-wave early exit ----
    #pragma unroll 1
    for (int it = 1; it < MAX_ITER; ++it) {
        v8f acc[8];
        #pragma unroll
        for (int n = 0; n < 8; ++n) acc[n] = xb[n];             // C starts at x + b

        const float* ap = zsh + lmod * LDS_STRIDE + 2 * lhalf;  // A-frag base (z)
        const float* wp = wsh + lmod * LDS_STRIDE + 2 * lhalf;  // B-frag base (W)
        #pragma unroll
        for (int t = 0; t < 32; ++t) {                          // K = 4*t .. 4*t+3
            const v2f a = *(const v2f*)(ap + t * 4);
            #pragma unroll
            for (int n = 0; n < 8; ++n) {                       // output cols 16n..16n+15
                v2f bf = *(const v2f*)(wp + t * 4 + n * 16 * LDS_STRIDE);
                acc[n] = __builtin_amdgcn_wmma_f32_16x16x4_f32(
                    false, a, false, bf, (short)0, acc[n], false, false);
            }
        }

        // z_next = tanh(acc); diff vs old z (same-lane reload), store back
        float maxd = 0.f;
        #pragma unroll
        for (int n = 0; n < 8; ++n) {
            const int col = n * 16 + lmod;
            #pragma unroll
            for (int m = 0; m < 8; ++m) {
                float* zp  = zsh + (m + 8 * lhalf) * LDS_STRIDE + col;
                float  old = *zp;                     // written by this same lane
                float  nw  = tanhf(acc[n][m]);
                maxd = fmaxf(maxd, fabsf(nw - old));
                *zp = nw;
            }
        }
        asm volatile("s_wait_dscnt 0x0" ::: "memory");  // publish z for A-frag reads

        bool active = (maxd > CONV_EPS);
#if __has_builtin(__builtin_amdgcn_ballot_w32)
        if (__builtin_amdgcn_ballot_w32(active) == 0u) break;   // wave-uniform exit
#else
        if (!__any(active)) break;
#endif
    }

    // ---- coalesced write-out from LDS (16 rows x 32 float4) ----
    {
        float* o = out + row0 * HIDDEN;
        #pragma unroll
        for (int p = 0; p < 16; ++p) {
            const int idx = p * 32 + lane;
            const int row = idx >> 5;
            const int c4  = idx & 31;
            float4 v = *(const float4*)(zsh + row * LDS_STRIDE + c4 * 4);
            *(float4*)(o + row * HIDDEN + c4 * 4) = v;
        }
    }
}

extern "C" void kernel_launch(void* const* d_in, const int* in_sizes, int n_in,
                              void* d_out, int out_size, void* d_ws, size_t ws_size,
                              hipStream_t stream) {
    const float* x = (const float*)d_in[0];   // [262144, 128]
    const float* W = (const float*)d_in[1];   // [128, 128]
    const float* b = (const float*)d_in[2];   // [128]
    float* out     = (float*)d_out;           // [262144, 128]

    const int batch = in_sizes[0] / HIDDEN;                   // 262144
    const int rows_per_block = ROWS_PER_WAVE * WAVES_PER_BLK; // 128
    const int grid = batch / rows_per_block;                  // 2048
    const size_t lds_bytes = (size_t)(W_LDS_FLOATS + WAVES_PER_BLK * Z_LDS_FLOATS) * 4; // 135168

    deq_f32_wmma_kernel<<<grid, THREADS, lds_bytes, stream>>>(x, W, b, out);
}